// FFM_Layer_16930761081257
// MI455X (gfx1250) — compile-verified
//
#include <hip/hip_runtime.h>

#define N_DENSE   13
#define N_SPARSE  26
#define SPARSE_DIM 100
#define N_FIELD   39
#define KDIM      16
#define NCOL      624          // N_FIELD * KDIM
#define NFEAT_ROW (N_DENSE + N_SPARSE)   // 39 cols of x per row
#define LDS_STRIDE 656         // 624 padded; 656 % 64 == 16 spreads banks across rows
#define ROWS_PER_BLOCK 16

typedef float v8f __attribute__((ext_vector_type(8)));
typedef float v4f __attribute__((ext_vector_type(4)));
typedef float v2f __attribute__((ext_vector_type(2)));

__global__ __launch_bounds__(256) void ffm_fused_kernel(
    const float* __restrict__ x,      // (B, 39)
    const float* __restrict__ w,      // (2613,)
    const float* __restrict__ bias,   // (1,)
    const float* __restrict__ v,      // (2613, 39, 16)
    float* __restrict__ out)          // (B,)
{
    __shared__ __attribute__((aligned(16))) float lds_field[ROWS_PER_BLOCK * LDS_STRIDE];
    __shared__ __attribute__((aligned(16))) float lds_A[ROWS_PER_BLOCK * 16];
    __shared__ int   lds_idx[ROWS_PER_BLOCK * N_SPARSE];
    __shared__ float lds_red[ROWS_PER_BLOCK * 16];

    const int t  = threadIdx.x;
    const int g0 = blockIdx.x * ROWS_PER_BLOCK;

    // ---- stage dense A (zero-padded 13->16) and sparse indices ----
    {
        int row = t >> 4, dd = t & 15;
        float a = 0.0f;
        if (dd < N_DENSE) a = x[(g0 + row) * NFEAT_ROW + dd];
        lds_A[row * 16 + dd] = a;
    }
    for (int u = t; u < ROWS_PER_BLOCK * N_SPARSE; u += 256) {
        int row = u / N_SPARSE, j = u % N_SPARSE;
        lds_idx[u] = (int)x[(g0 + row) * NFEAT_ROW + N_DENSE + j];
    }
    __syncthreads();

    // ---- sparse part: 26 gathered embedding rows, accumulated in registers ----
    // 16 threads per row; thread (row, l16) owns cols {4*l16 + 64*i}, i = 0..9
    const int row = t >> 4;
    const int l16 = t & 15;
    v4f acc[10];
    #pragma unroll
    for (int i = 0; i < 10; ++i) acc[i] = 0.0f;

    #pragma unroll 2
    for (int j = 0; j < N_SPARSE; ++j) {
        int feat = N_DENSE + j * SPARSE_DIM + lds_idx[row * N_SPARSE + j];
        const float* vr = v + feat * NCOL;
        #pragma unroll
        for (int i = 0; i < 9; ++i) {
            int c = 4 * l16 + 64 * i;
            acc[i] += *(const v4f*)(vr + c);   // global_load_b128, coalesced per row
        }
        if (l16 < 12) {                         // tail chunk: 624 = 9*64 + 48
            acc[9] += *(const v4f*)(vr + 4 * l16 + 576);
        }
    }

    // ---- dense part via V_WMMA_F32_16X16X4_F32 ----
    // C(16x16) per field f: rows = 16 batch rows, cols = k (0..15).
    // A: lds_A (16x16, K padded with zeros beyond 13); B: v[d, f, k], d = K.
    {
        const int lane = t & 31;
        const int wid  = t >> 5;           // 8 waves
        const int m    = lane & 15;        // A row (M) == B/C column (N)
        const int h    = lane >> 4;        // half-wave select
        for (int f = wid; f < N_FIELD; f += 8) {
            v8f c = 0.0f;
            const float* vb = v + f * KDIM;
            #pragma unroll
            for (int s = 0; s < 4; ++s) {
                int k0 = 4 * s + 2 * h;    // VGPR0 holds K = k0, VGPR1 holds K = k0+1
                v2f a, b;
                a.x = lds_A[m * 16 + k0];
                a.y = lds_A[m * 16 + k0 + 1];
                b.x = vb[k0 * NCOL + m];
                b.y = vb[(k0 + 1) * NCOL + m];
                c = __builtin_amdgcn_wmma_f32_16x16x4_f32(
                        false, a, false, b, (short)0, c, false, false);
            }
            #pragma unroll
            for (int r = 0; r < 8; ++r)    // C: VGPR r -> M = r + 8*h, N = m
                lds_field[(r + 8 * h) * LDS_STRIDE + f * KDIM + m] = c[r];
        }
    }
    __syncthreads();

    // ---- combine: add sparse register accumulators onto dense result in LDS ----
    #pragma unroll
    for (int i = 0; i < 9; ++i) {
        int c = 4 * l16 + 64 * i;
        v4f* p = (v4f*)&lds_field[row * LDS_STRIDE + c];
        *p += acc[i];
    }
    if (l16 < 12) {
        v4f* p = (v4f*)&lds_field[row * LDS_STRIDE + 4 * l16 + 576];
        *p += acc[9];
    }
    __syncthreads();

    // ---- reduce over fields: s_k, sq_k per (row, k) ----
    {
        int rr = t >> 4, k = t & 15;
        float s = 0.0f, q = 0.0f;
        #pragma unroll 1
        for (int f = 0; f < N_FIELD; ++f) {
            float val = lds_field[rr * LDS_STRIDE + f * KDIM + k];
            s += val;
            q += val * val;
        }
        lds_red[rr * 16 + k] = s * s - q;
    }
    __syncthreads();

    // ---- final: cross term + linear part + sigmoid ----
    if (t < ROWS_PER_BLOCK) {
        float cr = 0.0f;
        #pragma unroll
        for (int k = 0; k < 16; ++k) cr += lds_red[t * 16 + k];
        float lin = bias[0];
        const float* xr = x + (g0 + t) * NFEAT_ROW;
        #pragma unroll
        for (int d = 0; d < N_DENSE; ++d) lin += xr[d] * w[d];
        #pragma unroll 1
        for (int j = 0; j < N_SPARSE; ++j)
            lin += w[N_DENSE + j * SPARSE_DIM + lds_idx[t * N_SPARSE + j]];
        float z = lin + 0.5f * cr;
        out[g0 + t] = 1.0f / (1.0f + __expf(-z));
    }
}

extern "C" void kernel_launch(void* const* d_in, const int* in_sizes, int n_in,
                              void* d_out, int out_size, void* d_ws, size_t ws_size,
                              hipStream_t stream) {
    const float* x    = (const float*)d_in[0];
    const float* w    = (const float*)d_in[1];
    const float* bias = (const float*)d_in[2];
    const float* v    = (const float*)d_in[3];
    float* out = (float*)d_out;

    int nrows  = in_sizes[0] / NFEAT_ROW;          // 16384
    int blocks = nrows / ROWS_PER_BLOCK;           // 1024
    ffm_fused_kernel<<<blocks, 256, 0, stream>>>(x, w, bias, v, out);
}